// PygGCN_88072599371915
// MI455X (gfx1250) — compile-verified
//
#include <hip/hip_runtime.h>

typedef __attribute__((ext_vector_type(2))) float v2f;
typedef __attribute__((ext_vector_type(8))) float v8f;

#define HID 32   // both hidden and output feature width

// ---------------- utility kernels ----------------

__global__ void gcn_fill_f32(float* __restrict__ p, float v, long n) {
    long i = (long)blockIdx.x * blockDim.x + threadIdx.x;
    if (i < n) p[i] = v;
}

__global__ void gcn_degree(const int* __restrict__ dst, float* __restrict__ deg, int E) {
    int e = blockIdx.x * blockDim.x + threadIdx.x;
    if (e < E) atomicAdd(&deg[dst[e]], 1.0f);
}

__global__ void gcn_rsqrt_inplace(float* __restrict__ deg, int n) {
    int i = blockIdx.x * blockDim.x + threadIdx.x;
    if (i < n) deg[i] = rsqrtf(deg[i]);   // deg >= 1 always (self-loops)
}

// bias + ReLU, in place over [n_nodes, HID]
__global__ void gcn_bias_relu(float* __restrict__ io, const float* __restrict__ b, int n_nodes) {
    int gid  = blockIdx.x * blockDim.x + threadIdx.x;
    int node = gid >> 5;
    int f    = gid & 31;
    if (node < n_nodes) {
        long idx = (long)node * HID + f;
        io[idx] = fmaxf(io[idx] + b[f], 0.0f);
    }
}

// ---------------- WMMA GEMM: Out[nrows, 32] = A[nrows, K] @ W[K, 32] ----------------
// One wave per 16-row M-tile; two 16x16 f32 accumulators cover all 32 output cols.
// f32 WMMA 16x16x4 keeps full fp32 precision (reference is fp32).
__global__ __launch_bounds__(256) void gcn_gemm_wmma(const float* __restrict__ A,
                                                     const float* __restrict__ W,
                                                     float* __restrict__ Out,
                                                     int nrows, int K) {
    int wave = (int)((blockIdx.x * (long)blockDim.x + threadIdx.x) >> 5);
    int lane = threadIdx.x & 31;
    int m0   = wave * 16;
    if (m0 >= nrows) return;            // wave-uniform: EXEC stays all-ones for WMMA

    int half = lane >> 4;               // 0: K+{0,1} / rows M+0..7 ; 1: K+{2,3} / rows M+8..15
    int l    = lane & 15;
    int arow = m0 + l;
    if (arow >= nrows) arow = nrows - 1;          // clamp (tiles are exact for N=100000)
    const float* __restrict__ arow_ptr = A + (long)arow * K;

    v8f acc0 = {};                      // cols  0..15
    v8f acc1 = {};                      // cols 16..31

    for (int k0 = 0; k0 < K; k0 += 4) {
        int ka = k0 + half * 2;
        v2f a;
        a.x = arow_ptr[ka];
        a.y = arow_ptr[ka + 1];

        const float* __restrict__ wrow = W + (long)ka * HID;   // row ka of W (HID-wide)
        v2f b0, b1;
        b0.x = wrow[l];                 // B[ka  ][l]
        b0.y = wrow[HID + l];           // B[ka+1][l]
        b1.x = wrow[16 + l];            // B[ka  ][16+l]
        b1.y = wrow[HID + 16 + l];      // B[ka+1][16+l]

        acc0 = __builtin_amdgcn_wmma_f32_16x16x4_f32(false, a, false, b0,
                                                     (short)0, acc0, false, false);
        acc1 = __builtin_amdgcn_wmma_f32_16x16x4_f32(false, a, false, b1,
                                                     (short)0, acc1, false, false);
    }

    // C/D layout: acc[i] -> row m0 + i + half*8, col = l (acc0) / 16+l (acc1)
    for (int i = 0; i < 8; ++i) {
        int row = m0 + i + half * 8;
        if (row < nrows) {
            Out[(long)row * HID + l]      = acc0[i];
            Out[(long)row * HID + 16 + l] = acc1[i];
        }
    }
}

// ---------------- edge aggregation (gather * norm, atomic scatter-add) ----------------
// One wave per edge: 32 lanes = one contiguous 128B feature row (coalesced L2 traffic).
// Edges E..E+N-1 are the self-loops (src = dst = e - E).
__global__ void gcn_aggregate(const float* __restrict__ h,
                              const int* __restrict__ src,
                              const int* __restrict__ dst,
                              const float* __restrict__ dinv,
                              float* __restrict__ out,
                              int E, int N) {
    long gid = (long)blockIdx.x * blockDim.x + threadIdx.x;
    long e   = gid >> 5;
    int  f   = (int)(gid & 31);
    long total = (long)E + N;
    if (e >= total) return;

    int s, d;
    if (e < E) { s = src[e]; d = dst[e]; }
    else       { s = d = (int)(e - E); }

    float norm = dinv[s] * dinv[d];
    float v = h[(long)s * HID + f] * norm;
    atomicAdd(&out[(long)d * HID + f], v);
}

// ---------------- host-side launch ----------------

extern "C" void kernel_launch(void* const* d_in, const int* in_sizes, int n_in,
                              void* d_out, int out_size, void* d_ws, size_t ws_size,
                              hipStream_t stream) {
    const float* x   = (const float*)d_in[0];   // [N, 128]
    const int*   ei  = (const int*)d_in[1];     // [2, E] flattened
    const float* W1  = (const float*)d_in[2];   // [128, 32]
    const float* b1  = (const float*)d_in[3];   // [32]
    const float* W2  = (const float*)d_in[4];   // [32, 32]
    const float* b2  = (const float*)d_in[5];   // [32]

    const int FIN = 128;
    const int N = in_sizes[0] / FIN;
    const int E = in_sizes[1] / 2;
    const int* src = ei;
    const int* dst = ei + E;

    // workspace layout: deg[N] | h[N*HID] | agg1[N*HID]
    float* deg  = (float*)d_ws;
    float* h    = deg + N;
    float* agg1 = h + (long)N * HID;
    float* out  = (float*)d_out;

    const int BT = 256;
    long nfeat      = (long)N * HID;
    long agg_thr    = ((long)E + N) * 32;
    int  agg_blocks = (int)((agg_thr + BT - 1) / BT);
    int  gemm_waves = (N + 15) / 16;
    int  gemm_blk   = (gemm_waves * 32 + BT - 1) / BT;

    // 1) deg = 1 (self-loop) then count edge targets, then rsqrt in place
    gcn_fill_f32<<<(int)((N + BT - 1) / BT), BT, 0, stream>>>(deg, 1.0f, N);
    gcn_degree<<<(E + BT - 1) / BT, BT, 0, stream>>>(dst, deg, E);
    gcn_rsqrt_inplace<<<(N + BT - 1) / BT, BT, 0, stream>>>(deg, N);

    // 2) layer 1: h = x @ W1 (WMMA, K=128); aggregate into agg1; +b1, ReLU
    gcn_fill_f32<<<(int)((nfeat + BT - 1) / BT), BT, 0, stream>>>(agg1, 0.0f, nfeat);
    gcn_gemm_wmma<<<gemm_blk, BT, 0, stream>>>(x, W1, h, N, FIN);
    gcn_aggregate<<<agg_blocks, BT, 0, stream>>>(h, src, dst, deg, agg1, E, N);
    gcn_bias_relu<<<(int)((nfeat + BT - 1) / BT), BT, 0, stream>>>(agg1, b1, N);

    // 3) layer 2: h = agg1 @ W2 (WMMA, K=32); aggregate into d_out; +b2, ReLU
    gcn_gemm_wmma<<<gemm_blk, BT, 0, stream>>>(agg1, W2, h, N, HID);
    gcn_fill_f32<<<(int)((nfeat + BT - 1) / BT), BT, 0, stream>>>(out, 0.0f, nfeat);
    gcn_aggregate<<<agg_blocks, BT, 0, stream>>>(h, src, dst, deg, out, E, N);
    gcn_bias_relu<<<(int)((nfeat + BT - 1) / BT), BT, 0, stream>>>(out, b2, N);
}